// RIME_43319040147993
// MI455X (gfx1250) — compile-verified
//
#include <hip/hip_runtime.h>
#include <math.h>

typedef __attribute__((ext_vector_type(2))) float v2f;
typedef __attribute__((ext_vector_type(8))) float v8f;

namespace {
constexpr int NTIMES = 16;
constexpr int NFREQS = 32;
constexpr int NSRC   = 1024;
constexpr int NBL    = 28;
constexpr float C_MPS = 299792458.0f;
}

// One block per (baseline, time). Two wave32s per block; wave w owns freqs
// [w*16, w*16+16). Each wave reduces 1024 sources via chained
// V_WMMA_F32_16X16X4_F32 against an all-ones B (row-sum trick), full f32.
//
// Trig: v_sin_f32/v_cos_f32 take "revolutions" (sin(2*pi*x)), so we fold the
// -2*pi factor of the fringe phase into the hardware op:
//   rev = (-f/c) * (bvec . shat)   ->   sin(2*pi*rev) == sin(phase)
// |rev| <= ~100, well inside the HW range; no libm slow path, no branches.
__global__ __launch_bounds__(64) void rime_vis_kernel(
    const float* __restrict__ sky,       // [NFREQS][NSRC]
    const float* __restrict__ beam,      // [NMODEL][NTIMES][NFREQS][NSRC]
    const float* __restrict__ antpos,    // [NANTS][3]
    const float* __restrict__ src_dirs,  // [NTIMES][NSRC][3]
    const float* __restrict__ freqs,     // [NFREQS]
    const int*   __restrict__ ant2model, // [NANTS]
    const int*   __restrict__ bls,       // [NBL][2]
    float*       __restrict__ out)       // [NBL][NTIMES][NFREQS][2] (re,im)
{
    const int tile = blockIdx.x;          // 0 .. NBL*NTIMES-1
    const int b = tile / NTIMES;
    const int t = tile - b * NTIMES;

    const int wave = threadIdx.x >> 5;    // freq half: 0 or 1
    const int lane = threadIdx.x & 31;
    const int m    = lane & 15;           // A-matrix row = freq within half
    const int kb   = (lane >> 4) << 1;    // K pair base: 0 (lanes 0-15), 2 (lanes 16-31)

    // Baseline geometry + beam model selection (uniform per block).
    const int a1 = bls[2 * b + 0];
    const int a2 = bls[2 * b + 1];
    const float bx = antpos[3 * a1 + 0] - antpos[3 * a2 + 0];
    const float by = antpos[3 * a1 + 1] - antpos[3 * a2 + 1];
    const float bz = antpos[3 * a1 + 2] - antpos[3 * a2 + 2];
    const int m1 = ant2model[a1];
    const int m2 = ant2model[a2];

    // Per-lane frequency row; fringe coefficient in *revolutions* per (b.s).
    const int f = wave * 16 + m;
    const float wrev = -freqs[f] / C_MPS;

    const float* __restrict__ b1row  = beam + (((size_t)m1 * NTIMES + t) * NFREQS + f) * NSRC;
    const float* __restrict__ b2row  = beam + (((size_t)m2 * NTIMES + t) * NFREQS + f) * NSRC;
    const float* __restrict__ skyrow = sky + (size_t)f * NSRC;
    const float* __restrict__ sd     = src_dirs + (size_t)t * NSRC * 3;

    v8f cre = {};   // f32 accumulator, real part
    v8f cim = {};   // f32 accumulator, imag part
    const v2f ones = {1.0f, 1.0f};  // B (4x16) all ones -> D columns = row sums

#pragma unroll 2
    for (int s0 = 0; s0 < NSRC; s0 += 4) {
        const int sbase = s0 + kb;      // this lane's two consecutive sources

        // Vector loads: the lane's K-pair is contiguous in s.
        const v2f bm1 = *(const v2f*)(b1row + sbase);
        const v2f bm2 = *(const v2f*)(b2row + sbase);
        const v2f sk  = *(const v2f*)(skyrow + sbase);

        v2f are, aim;
#pragma unroll
        for (int k = 0; k < 2; ++k) {
            const int s = sbase + k;
            const float sx = sd[3 * s + 0];
            const float sy = sd[3 * s + 1];
            const float sz = sd[3 * s + 2];
            const float rev = wrev * (bx * sx + by * sy + bz * sz);
            const float cs = __builtin_amdgcn_cosf(rev);  // cos(2*pi*rev)
            const float sn = __builtin_amdgcn_sinf(rev);  // sin(2*pi*rev)
            const float p = bm1[k] * bm2[k] * sk[k];
            are[k] = p * cs;
            aim[k] = p * sn;
        }
        // D = A(16x4) * ones(4x16) + C : accumulates 4 sources into every column.
        cre = __builtin_amdgcn_wmma_f32_16x16x4_f32(
            false, are, false, ones, (short)0, cre, false, false);
        cim = __builtin_amdgcn_wmma_f32_16x16x4_f32(
            false, aim, false, ones, (short)0, cim, false, false);
    }

    // C/D 16x16 f32 layout: VGPR i -> row i (lanes 0-15) / row 8+i (lanes 16-31);
    // all 16 columns are identical (row sums), so lanes 0 and 16 write results.
    if (m == 0) {
        const int rowbase = wave * 16 + ((lane >> 4) ? 8 : 0);
#pragma unroll
        for (int i = 0; i < 8; ++i) {
            const int ff = rowbase + i;
            const size_t o = (((size_t)b * NTIMES + t) * NFREQS + ff) * 2;
            out[o + 0] = cre[i];
            out[o + 1] = cim[i];
        }
    }
}

extern "C" void kernel_launch(void* const* d_in, const int* in_sizes, int n_in,
                              void* d_out, int out_size, void* d_ws, size_t ws_size,
                              hipStream_t stream) {
    const float* sky       = (const float*)d_in[0];
    const float* beam      = (const float*)d_in[1];
    const float* antpos    = (const float*)d_in[2];
    const float* src_dirs  = (const float*)d_in[3];
    const float* freqs     = (const float*)d_in[4];
    const int*   ant2model = (const int*)d_in[5];
    const int*   bls       = (const int*)d_in[6];
    float*       out       = (float*)d_out;

    (void)in_sizes; (void)n_in; (void)out_size; (void)d_ws; (void)ws_size;

    rime_vis_kernel<<<NBL * NTIMES, 64, 0, stream>>>(
        sky, beam, antpos, src_dirs, freqs, ant2model, bls, out);
}